// PolicyNet_5360119185793
// MI455X (gfx1250) — compile-verified
//
#include <hip/hip_runtime.h>
#include <cstdint>
#include <cstddef>

typedef __attribute__((ext_vector_type(2))) float v2f;
typedef __attribute__((ext_vector_type(4))) float v4f;
typedef __attribute__((ext_vector_type(8))) float v8f;

#define B_TOT  4096
#define N_KEYS 200
#define DIM    128
#define NWORDS 7          // ceil(200/32)
#define C_CLIP 10.0f

// 16 batch rows per block, one wave32 per batch row in phases 1 & 3.
__global__ __launch_bounds__(512)
void policy_net_fused_kernel(const float* __restrict__ query,       // [B,1,D]
                             const float* __restrict__ gkey,        // [B,N,D]
                             const float* __restrict__ gval,        // [B,N,D]
                             const float* __restrict__ lkey,        // [B,N,D]
                             const float* __restrict__ wout,        // [D,D] row-major
                             const unsigned char* __restrict__ mask,// [B,N] bool
                             float* __restrict__ out)               // [B,1,N]
{
    __shared__ __align__(16) float s_glimpse[16][132]; // +4 pad: conflict-free WMMA A reads
    __shared__ __align__(16) float s_proj[16][128];
    __shared__ __align__(16) float s_logits[16][200];

    const int tid    = threadIdx.x;
    const int wave   = tid >> 5;
    const int lane   = tid & 31;
    const int bLocal = wave;
    const int b      = blockIdx.x * 16 + bLocal;

    // ---- Preload mask row once per wave: 7 coalesced byte loads -> SGPR bitmask
    const unsigned char* mrow = mask + (size_t)b * N_KEYS;
    uint32_t mbits[NWORDS];
    #pragma unroll
    for (int w = 0; w < NWORDS; ++w) {
        int n = w * 32 + lane;
        int pred = (n < N_KEYS) ? (mrow[n] != 0) : 0;
        mbits[w] = (uint32_t)__ballot(pred);   // wave-uniform
    }

    // ---------------- Phase 1: masked MHA glimpse, single streaming pass ----
    {
        // lane l owns dims [4l, 4l+3]; head h = l/4 owns dims [16h, 16h+15]
        const v4f q4 = *(const v4f*)(query + (size_t)b * DIM + 4 * lane);
        const float* krow = gkey + (size_t)b * N_KEYS * DIM + 4 * lane;
        const float* vrow = gval + (size_t)b * N_KEYS * DIM + 4 * lane;
        const float scale = 0.25f; // 1/sqrt(HD=16)

        float m = -INFINITY, s = 0.0f;
        float ax = 0.f, ay = 0.f, az = 0.f, aw = 0.f;

        #pragma unroll
        for (int w = 0; w < NWORDS; ++w) {
            uint32_t bits = (uint32_t)__builtin_amdgcn_readfirstlane(mbits[w]);
            while (bits) {                         // visit only unmasked rows
                const int i = __builtin_ctz(bits);
                bits &= bits - 1;
                const int n = w * 32 + i;
                v4f k4 = __builtin_nontemporal_load((const v4f*)(krow + (size_t)n * DIM));
                v4f v4 = __builtin_nontemporal_load((const v4f*)(vrow + (size_t)n * DIM));
                float p = q4.x * k4.x + q4.y * k4.y + q4.z * k4.z + q4.w * k4.w;
                p += __shfl_xor(p, 1, 32);         // reduce within 4-lane head group
                p += __shfl_xor(p, 2, 32);
                float sc = p * scale;
                float nm = fmaxf(m, sc);
                float corr = __expf(m - nm);       // m==-inf -> 0 on first valid row
                float e    = __expf(sc - nm);
                s  = s * corr + e;
                ax = ax * corr + e * v4.x;
                ay = ay * corr + e * v4.y;
                az = az * corr + e * v4.z;
                aw = aw * corr + e * v4.w;
                m  = nm;
            }
        }
        const float inv = 1.0f / s;
        s_glimpse[bLocal][4 * lane + 0] = ax * inv;
        s_glimpse[bLocal][4 * lane + 1] = ay * inv;
        s_glimpse[bLocal][4 * lane + 2] = az * inv;
        s_glimpse[bLocal][4 * lane + 3] = aw * inv;
    }
    __syncthreads();

    // ---------------- Phase 2: proj[16][128] = glimpse @ W_out^T (WMMA f32) --
    // proj[m][j] = sum_k glimpse[m][k] * wout[j][k]
    if (wave < 8) {
        const int tileN = wave * 16;         // output column tile
        const int mi = lane & 15;            // row within 16
        const int kh = lane >> 4;            // K half-select (0 -> K0/K1, 1 -> K2/K3)
        v8f acc = {};
        #pragma unroll 4
        for (int k0 = 0; k0 < DIM; k0 += 4) {
            v2f a, bf;
            // A 16x4 f32 frag: VGPR0 = K(2*kh), VGPR1 = K(2*kh+1), rows in lanes
            a.x = s_glimpse[mi][k0 + 2 * kh];
            a.y = s_glimpse[mi][k0 + 2 * kh + 1];
            // B 4x16 frag: row K = k0+2*kh(+1), col j = tileN+mi; B[k][j] = wout[j][k]
            const float* wp = wout + (size_t)(tileN + mi) * DIM + k0 + 2 * kh;
            v2f w2 = *(const v2f*)wp;         // 8B-aligned (k0+2*kh is even)
            bf.x = w2.x;
            bf.y = w2.y;
            acc = __builtin_amdgcn_wmma_f32_16x16x4_f32(
                false, a, false, bf, (short)0, acc, false, false);
        }
        // C/D layout: VGPR r -> row r (lanes 0-15) / row r+8 (lanes 16-31)
        #pragma unroll
        for (int r = 0; r < 8; ++r)
            s_proj[r + 8 * kh][tileN + mi] = acc[r];
    }
    __syncthreads();

    // ---------------- Phase 3: tanh-clipped logit attention + log_softmax ----
    {
        const v4f g4 = *(const v4f*)&s_proj[bLocal][4 * lane];
        const float* lrow = lkey + (size_t)b * N_KEYS * DIM + 4 * lane;
        const float scale = 0.08838834764831845f; // 1/sqrt(128)

        // pre-fill this batch row's logits with -inf (masked entries stay -inf)
        for (int n = lane; n < N_KEYS; n += 32)
            s_logits[bLocal][n] = -INFINITY;

        float M = -INFINITY, S = 0.0f;
        #pragma unroll
        for (int w = 0; w < NWORDS; ++w) {
            uint32_t bits = (uint32_t)__builtin_amdgcn_readfirstlane(mbits[w]);
            while (bits) {
                const int i = __builtin_ctz(bits);
                bits &= bits - 1;
                const int n = w * 32 + i;
                v4f k4 = __builtin_nontemporal_load((const v4f*)(lrow + (size_t)n * DIM));
                float p = g4.x * k4.x + g4.y * k4.y + g4.z * k4.z + g4.w * k4.w;
                p += __shfl_xor(p, 1, 32);         // full 32-lane reduction
                p += __shfl_xor(p, 2, 32);
                p += __shfl_xor(p, 4, 32);
                p += __shfl_xor(p, 8, 32);
                p += __shfl_xor(p, 16, 32);
                const float logit = C_CLIP * tanhf(p * scale);
                // online log-sum-exp (identical in every lane)
                float nm = fmaxf(M, logit);
                S = S * __expf(M - nm) + __expf(logit - nm);
                M = nm;
                if (lane == 0) s_logits[bLocal][n] = logit;
            }
        }
        __syncthreads();  // make lane-0 LDS writes visible (uniform for all waves)

        const float logS = __logf(S);
        float* orow = out + (size_t)b * N_KEYS;
        for (int n = lane; n < N_KEYS; n += 32) {
            float lg = s_logits[bLocal][n];           // -inf stays -inf
            orow[n] = lg - M - logS;
        }
    }
}

extern "C" void kernel_launch(void* const* d_in, const int* in_sizes, int n_in,
                              void* d_out, int out_size, void* d_ws, size_t ws_size,
                              hipStream_t stream) {
    (void)in_sizes; (void)n_in; (void)d_ws; (void)ws_size; (void)out_size;
    const float* query = (const float*)d_in[0];
    const float* gkey  = (const float*)d_in[1];
    const float* gval  = (const float*)d_in[2];
    const float* lkey  = (const float*)d_in[3];
    const float* wout  = (const float*)d_in[4];
    const unsigned char* mask = (const unsigned char*)d_in[5];
    float* out = (float*)d_out;

    dim3 grid(B_TOT / 16);   // 256 blocks, 16 batch rows each
    dim3 block(512);         // 16 wave32 waves
    policy_net_fused_kernel<<<grid, block, 0, stream>>>(
        query, gkey, gval, lkey, wout, mask, out);
}